// PhaseClusterAdapter_69346541961554
// MI455X (gfx1250) — compile-verified
//
#include <hip/hip_runtime.h>
#include <hip/hip_bf16.h>
#include <math.h>

typedef __attribute__((ext_vector_type(2))) float v2f;
typedef __attribute__((ext_vector_type(8))) float v8f;

#define N_B    16
#define C_DIM  256
#define S_DIM  30
#define P_DIM  176            // H*W = 16*11
#define NCOL   2816           // N_B * P_DIM
#define KC     1024           // 4 * C_DIM

__device__ __forceinline__ float waveReduceSum(float v) {
#pragma unroll
  for (int m = 16; m >= 1; m >>= 1) v += __shfl_xor(v, m, 32);
  return v;
}

// ---------------------------------------------------------------------------
// Kernel A: one wave per (nn, p) column.
//   - normalize 4 prototypes of length 256
//   - per s: ||x||, 4 cosine sims, softmax/T, argmax (hard assign),
//     atomic accumulate soft sums (for cluster_indices),
//     running max of (assigned ? x : 0) per cluster  == reference max-pool
//   - write clustered[kc][col] row-major (kc = k*256 + c, col = nn*176 + p)
// ---------------------------------------------------------------------------
__global__ void __launch_bounds__(32)
cluster_kernel(const float* __restrict__ x, const float* __restrict__ protos,
               const float* __restrict__ tptr, float* __restrict__ cl,
               float* __restrict__ accum)
{
  const int b    = blockIdx.x;         // == column index
  const int nn   = b / P_DIM;
  const int p    = b % P_DIM;
  const int lane = threadIdx.x;
  const float tinv = 1.0f / tptr[0];

  // normalized prototypes, 8 channels per lane per cluster
  float pk[4][8];
#pragma unroll
  for (int k = 0; k < 4; ++k) {
    float ss = 0.f;
#pragma unroll
    for (int j = 0; j < 8; ++j) {
      float v = protos[(p * 4 + k) * C_DIM + j * 32 + lane];
      pk[k][j] = v; ss += v * v;
    }
    ss = waveReduceSum(ss);
    float inv = 1.0f / fmaxf(sqrtf(ss), 1e-12f);
#pragma unroll
    for (int j = 0; j < 8; ++j) pk[k][j] *= inv;
  }

  float mx[4][8];
#pragma unroll
  for (int k = 0; k < 4; ++k)
#pragma unroll
    for (int j = 0; j < 8; ++j) mx[k][j] = -INFINITY;

  for (int s = 0; s < S_DIM; ++s) {
    const int base = (nn * C_DIM * S_DIM + s) * P_DIM + p;   // c stride = S*P
    float xv[8];
    float ssx = 0.f;
#pragma unroll
    for (int j = 0; j < 8; ++j) {
      float v = x[base + (j * 32 + lane) * (S_DIM * P_DIM)];
      xv[j] = v; ssx += v * v;
    }
    float d0 = 0.f, d1 = 0.f, d2 = 0.f, d3 = 0.f;
#pragma unroll
    for (int j = 0; j < 8; ++j) {
      d0 += xv[j] * pk[0][j]; d1 += xv[j] * pk[1][j];
      d2 += xv[j] * pk[2][j]; d3 += xv[j] * pk[3][j];
    }
    ssx = waveReduceSum(ssx);
    d0 = waveReduceSum(d0); d1 = waveReduceSum(d1);
    d2 = waveReduceSum(d2); d3 = waveReduceSum(d3);
    const float inv = 1.0f / fmaxf(sqrtf(ssx), 1e-12f);
    float sim[4] = { d0 * inv, d1 * inv, d2 * inv, d3 * inv };

    int kb = 0;
#pragma unroll
    for (int k = 1; k < 4; ++k) if (sim[k] > sim[kb]) kb = k;   // first-max ties

    // softmax(sim / T)
    float l[4], lm = -INFINITY;
#pragma unroll
    for (int k = 0; k < 4; ++k) { l[k] = sim[k] * tinv; lm = fmaxf(lm, l[k]); }
    float e[4], es = 0.f;
#pragma unroll
    for (int k = 0; k < 4; ++k) { e[k] = expf(l[k] - lm); es += e[k]; }
    const float esr = 1.0f / es;
    if (lane == 0) {
#pragma unroll
      for (int k = 0; k < 4; ++k)
        atomicAdd(&accum[(nn * S_DIM + s) * 4 + k], e[k] * esr);
    }
#pragma unroll
    for (int k = 0; k < 4; ++k)
#pragma unroll
      for (int j = 0; j < 8; ++j) {
        float cand = (k == kb) ? xv[j] : 0.0f;   // 0 products participate in max
        mx[k][j] = fmaxf(mx[k][j], cand);
      }
  }

#pragma unroll
  for (int k = 0; k < 4; ++k)
#pragma unroll
    for (int j = 0; j < 8; ++j)
      cl[(k * C_DIM + j * 32 + lane) * NCOL + b] = mx[k][j];
}

// ---------------------------------------------------------------------------
// WMMA f32 16x16x4 GEMMs.  A: weights [256 x KDIM] row-major.  B: acts
// [KDIM x 2816] row-major.  Each wave: one 16-row x 64-col tile, 4 D accums
// reusing a single A load per k-step.
//  A lane layout: row = lane%16, k-pair = (lane<16) ? {0,1} : {2,3}
//  B lane layout: col = lane%16, k-pair = (lane<16) ? {0,1} : {2,3}
//  D lane layout: col = lane%16, rows = vgpr + 8*(lane>=16)
// ---------------------------------------------------------------------------
__global__ void __launch_bounds__(256)
gemm1_kernel(const float* __restrict__ w1, const float* __restrict__ cl,
             float* __restrict__ y1)
{
  const int wid  = (blockIdx.x * 256 + threadIdx.x) >> 5;
  const int lane = threadIdx.x & 31;
  const int mt = wid & 15, nq = wid >> 4;         // 16 m-tiles x 44 n-quads
  const int mbase = mt * 16, nbase = nq * 64;
  const int hl = lane >> 4, l16 = lane & 15;
  const int row = mbase + l16, koff = hl * 2;

  v8f acc0 = {0.f,0.f,0.f,0.f,0.f,0.f,0.f,0.f};
  v8f acc1 = acc0, acc2 = acc0, acc3 = acc0;

  for (int kb = 0; kb < KC; kb += 4) {
    v2f a = *(const v2f*)(w1 + row * KC + kb + koff);
    const float* br0 = cl + (kb + koff) * NCOL + nbase + l16;
    const float* br1 = br0 + NCOL;
    v2f b0 = { br0[0],  br1[0]  };
    v2f b1 = { br0[16], br1[16] };
    v2f b2 = { br0[32], br1[32] };
    v2f b3 = { br0[48], br1[48] };
    acc0 = __builtin_amdgcn_wmma_f32_16x16x4_f32(false, a, false, b0, (short)0, acc0, false, false);
    acc1 = __builtin_amdgcn_wmma_f32_16x16x4_f32(false, a, false, b1, (short)0, acc1, false, false);
    acc2 = __builtin_amdgcn_wmma_f32_16x16x4_f32(false, a, false, b2, (short)0, acc2, false, false);
    acc3 = __builtin_amdgcn_wmma_f32_16x16x4_f32(false, a, false, b3, (short)0, acc3, false, false);
  }
#pragma unroll
  for (int v = 0; v < 8; ++v) {
    const int o = mbase + v + 8 * hl;
    y1[o * NCOL + nbase + l16 +  0] = acc0[v];
    y1[o * NCOL + nbase + l16 + 16] = acc1[v];
    y1[o * NCOL + nbase + l16 + 32] = acc2[v];
    y1[o * NCOL + nbase + l16 + 48] = acc3[v];
  }
}

__global__ void __launch_bounds__(256)
gemm2_kernel(const float* __restrict__ w2, const float* __restrict__ y2,
             const float* __restrict__ b2, float* __restrict__ out)
{
  const int wid  = (blockIdx.x * 256 + threadIdx.x) >> 5;
  const int lane = threadIdx.x & 31;
  const int mt = wid & 15, nq = wid >> 4;
  const int mbase = mt * 16, nbase = nq * 64;
  const int hl = lane >> 4, l16 = lane & 15;
  const int row = mbase + l16, koff = hl * 2;

  v8f acc0 = {0.f,0.f,0.f,0.f,0.f,0.f,0.f,0.f};
  v8f acc1 = acc0, acc2 = acc0, acc3 = acc0;

  for (int kb = 0; kb < C_DIM; kb += 4) {
    v2f a = *(const v2f*)(w2 + row * C_DIM + kb + koff);
    const float* br0 = y2 + (kb + koff) * NCOL + nbase + l16;
    const float* br1 = br0 + NCOL;
    v2f b0 = { br0[0],  br1[0]  };
    v2f b1 = { br0[16], br1[16] };
    v2f b2v = { br0[32], br1[32] };
    v2f b3 = { br0[48], br1[48] };
    acc0 = __builtin_amdgcn_wmma_f32_16x16x4_f32(false, a, false, b0, (short)0, acc0, false, false);
    acc1 = __builtin_amdgcn_wmma_f32_16x16x4_f32(false, a, false, b1, (short)0, acc1, false, false);
    acc2 = __builtin_amdgcn_wmma_f32_16x16x4_f32(false, a, false, b2v, (short)0, acc2, false, false);
    acc3 = __builtin_amdgcn_wmma_f32_16x16x4_f32(false, a, false, b3, (short)0, acc3, false, false);
  }
  // out[nn, o, p]: col = nn*176 + p ; 176 % 16 == 0 so each 16-tile stays in one nn
#pragma unroll
  for (int v = 0; v < 8; ++v) {
    const int o = mbase + v + 8 * hl;
    const float bias = b2[o];
#pragma unroll
    for (int t = 0; t < 4; ++t) {
      const int col = nbase + t * 16 + l16;
      const int nn = col / P_DIM, pp = col % P_DIM;
      float val = (t == 0 ? acc0[v] : t == 1 ? acc1[v] : t == 2 ? acc2[v] : acc3[v]);
      out[nn * (C_DIM * P_DIM) + o * P_DIM + pp] = val + bias;
    }
  }
}

// ---------------------------------------------------------------------------
// BatchNorm training-mode stats per channel (biased var over 2816 samples)
// ---------------------------------------------------------------------------
__global__ void __launch_bounds__(256)
bnstats_kernel(const float* __restrict__ y1, float* __restrict__ mean,
               float* __restrict__ istd)
{
  __shared__ float ssum[256], ssq[256];
  const int o = blockIdx.x, t = threadIdx.x;
  float s = 0.f, q = 0.f;
  for (int c = t; c < NCOL; c += 256) {
    float v = y1[o * NCOL + c];
    s += v; q += v * v;
  }
  ssum[t] = s; ssq[t] = q;
  __syncthreads();
  for (int off = 128; off > 0; off >>= 1) {
    if (t < off) { ssum[t] += ssum[t + off]; ssq[t] += ssq[t + off]; }
    __syncthreads();
  }
  if (t == 0) {
    float m = ssum[0] / (float)NCOL;
    float var = ssq[0] / (float)NCOL - m * m;
    mean[o] = m;
    istd[o] = rsqrtf(var + 1e-5f);
  }
}

__global__ void __launch_bounds__(256)
bngelu_kernel(const float* __restrict__ y1, const float* __restrict__ mean,
              const float* __restrict__ istd, const float* __restrict__ gamma,
              const float* __restrict__ beta, float* __restrict__ y2)
{
  const int idx = blockIdx.x * 256 + threadIdx.x;     // exactly 256*2816 threads
  const int o = idx / NCOL;
  float v = (y1[idx] - mean[o]) * istd[o] * gamma[o] + beta[o];
  y2[idx] = 0.5f * v * (1.0f + erff(v * 0.70710678118654752f));   // exact GELU
}

__global__ void cidx_kernel(const float* __restrict__ accum, float* __restrict__ outIdx)
{
  const int i = blockIdx.x * blockDim.x + threadIdx.x;
  if (i >= N_B * S_DIM) return;
  const float* a = accum + i * 4;
  int kb = 0; float bv = a[0];
#pragma unroll
  for (int k = 1; k < 4; ++k) if (a[k] > bv) { bv = a[k]; kb = k; }
  outIdx[i] = (float)kb;
}

extern "C" void kernel_launch(void* const* d_in, const int* in_sizes, int n_in,
                              void* d_out, int out_size, void* d_ws, size_t ws_size,
                              hipStream_t stream)
{
  const float* x      = (const float*)d_in[0];
  const float* protos = (const float*)d_in[1];
  const float* temp   = (const float*)d_in[2];
  const float* w1     = (const float*)d_in[3];
  const float* gamma  = (const float*)d_in[4];
  const float* beta   = (const float*)d_in[5];
  const float* w2     = (const float*)d_in[6];
  const float* b2     = (const float*)d_in[7];
  float* out = (float*)d_out;

  float* ws    = (float*)d_ws;
  float* cl    = ws;                       // 1024 * 2816
  float* y1    = cl + (size_t)KC * NCOL;   //  256 * 2816
  float* y2    = y1 + (size_t)C_DIM * NCOL;
  float* mean  = y2 + (size_t)C_DIM * NCOL;
  float* istd  = mean + C_DIM;
  float* accum = istd + C_DIM;             // 16*30*4 atomic soft sums

  hipMemsetAsync(accum, 0, (size_t)N_B * S_DIM * 4 * sizeof(float), stream);
  cluster_kernel<<<NCOL, 32, 0, stream>>>(x, protos, temp, cl, accum);
  gemm1_kernel<<<88, 256, 0, stream>>>(w1, cl, y1);            // 704 waves
  bnstats_kernel<<<C_DIM, 256, 0, stream>>>(y1, mean, istd);
  bngelu_kernel<<<2816, 256, 0, stream>>>(y1, mean, istd, gamma, beta, y2);
  gemm2_kernel<<<88, 256, 0, stream>>>(w2, y2, b2, out);       // 704 waves
  cidx_kernel<<<2, 256, 0, stream>>>(accum, out + (size_t)N_B * C_DIM * P_DIM);
}